// DHCF_79774722556261
// MI455X (gfx1250) — compile-verified
//
#include <hip/hip_runtime.h>

#define NU 100000
#define NI 50000
#define DD 128
#define NNZ_G 2000000
#define NNZ_M 1000000
#define BQ 8192

typedef __attribute__((ext_vector_type(2))) float v2f;
typedef __attribute__((ext_vector_type(8))) float v8f;

__device__ __forceinline__ void atomAddF(float* p, float v) {
  __hip_atomic_fetch_add(p, v, __ATOMIC_RELAXED, __HIP_MEMORY_SCOPE_AGENT);
}

// ---------------------------------------------------------------------------
// Kernel 1: A = concat(user_table, item_table)   (float4-vectorized copy)
// A is the running accumulator: folds the "+ table" term of the combine.
// ---------------------------------------------------------------------------
__global__ void init_acc(const float4* __restrict__ ut,
                         const float4* __restrict__ it,
                         float4* __restrict__ A) {
  long i = (long)blockIdx.x * blockDim.x + threadIdx.x;
  const long NU4 = (long)NU * (DD / 4);
  const long NT4 = (long)(NU + NI) * (DD / 4);
  if (i < NU4) {
    A[i] = ut[i];
  } else if (i < NT4) {
    A[i] = it[i - NU4];
  }
}

// ---------------------------------------------------------------------------
// Kernel 2: graph SpMM scatter over concatenated embedding:
//   A[r] += v * src[c]   where src = user_table (c<NU) else item_table (c-NU)
// One 32-lane group per edge, float4 per lane -> 128 dims. Atomics hit L2
// (accumulator + tables are L2-resident: ~154MB < 192MB global L2).
// ---------------------------------------------------------------------------
__global__ void scatter_g(const int* __restrict__ rows,
                          const int* __restrict__ cols,
                          const float* __restrict__ vals,
                          const float* __restrict__ ut,
                          const float* __restrict__ it,
                          float* __restrict__ A, int nnz) {
  long t = (long)blockIdx.x * blockDim.x + threadIdx.x;
  int e = (int)(t >> 5);
  int lane = (int)(t & 31);
  if (e >= nnz) return;
  int r = rows[e];
  int c = cols[e];
  float v = vals[e];
  const float* src = (c < NU) ? (ut + (long)c * DD) : (it + (long)(c - NU) * DD);
  float4 x = *(const float4*)(src + lane * 4);
  float* dst = A + (long)r * DD + lane * 4;
  atomAddF(dst + 0, v * x.x);
  atomAddF(dst + 1, v * x.y);
  atomAddF(dst + 2, v * x.z);
  atomAddF(dst + 3, v * x.w);
}

// ---------------------------------------------------------------------------
// Kernel 3: homogeneous SpMM scatter:  A[row_off + r] += v * tbl[c]
// (used for m1 with row_off=0 and m2 with row_off=NU)
// ---------------------------------------------------------------------------
__global__ void scatter_m(const int* __restrict__ rows,
                          const int* __restrict__ cols,
                          const float* __restrict__ vals,
                          const float* __restrict__ tbl,
                          float* __restrict__ A, long row_off, int nnz) {
  long t = (long)blockIdx.x * blockDim.x + threadIdx.x;
  int e = (int)(t >> 5);
  int lane = (int)(t & 31);
  if (e >= nnz) return;
  int r = rows[e];
  int c = cols[e];
  float v = vals[e];
  float4 x = *(const float4*)(tbl + (long)c * DD + lane * 4);
  float* dst = A + (row_off + r) * DD + lane * 4;
  atomAddF(dst + 0, v * x.x);
  atomAddF(dst + 1, v * x.y);
  atomAddF(dst + 2, v * x.z);
  atomAddF(dst + 3, v * x.w);
}

// ---------------------------------------------------------------------------
// Kernel 4: batched pair dot via V_WMMA_F32_16X16X4_F32 (full f32 precision).
// One wave (32 lanes) per block handles 16 (user,item) pairs.
//   A-tile: 16x4 slice of 16 user rows     (lane m holds K=koff..koff+1)
//   B-tile: 4x16 slice of 16 item rows^T   (same per-lane addressing)
//   D = A x B^T accumulated over K=128 in 32 WMMA steps; gamma = diag(D)/9.
// 16x16x4 f32 A/B layout: VGPR0/1 = K={0,1} on lanes 0-15, K={2,3} on 16-31.
// D layout: diag(p) = c[p] on lane p (p<8), c[p-8] on lane p+16 (p>=8).
// ---------------------------------------------------------------------------
__global__ void wmma_pair_dot(const float* __restrict__ A,
                              const int* __restrict__ users,
                              const int* __restrict__ items,
                              float* __restrict__ out) {
  const int lane = threadIdx.x;        // 0..31, all lanes active (EXEC all-1s)
  const int base = blockIdx.x * 16;    // 16 pairs per block
  const int m = lane & 15;             // which pair within the tile
  const int koff = (lane >> 4) * 2;    // lanes 0-15 -> K+{0,1}; 16-31 -> K+{2,3}

  const float* Urow = A + (long)users[base + m] * DD;
  const float* Irow = A + ((long)NU + items[base + m]) * DD;

  v8f c = {};
#pragma unroll
  for (int k = 0; k < DD; k += 4) {
    v2f a, b;
    a.x = Urow[k + koff];
    a.y = Urow[k + koff + 1];
    b.x = Irow[k + koff];
    b.y = Irow[k + koff + 1];
    // (neg_a, A, neg_b, B, c_mod, C, reuse_a, reuse_b)
    c = __builtin_amdgcn_wmma_f32_16x16x4_f32(false, a, false, b,
                                              (short)0, c, false, false);
  }

  const float s = 1.0f / 9.0f;  // (1/3)*(1/3) from the two combines
#pragma unroll
  for (int v = 0; v < 8; ++v) {
    float val = c[v] * s;
    if (lane == v)      out[base + v]     = val;  // diag p=v      (vgpr v, lane v)
    if (lane == v + 24) out[base + v + 8] = val;  // diag p=v+8    (vgpr v, lane v+24)
  }
}

// ---------------------------------------------------------------------------
extern "C" void kernel_launch(void* const* d_in, const int* in_sizes, int n_in,
                              void* d_out, int out_size, void* d_ws, size_t ws_size,
                              hipStream_t stream) {
  const float* user_table = (const float*)d_in[0];
  const float* item_table = (const float*)d_in[1];
  const float* g_vals  = (const float*)d_in[2];
  const float* m1_vals = (const float*)d_in[3];
  const float* m2_vals = (const float*)d_in[4];
  const int* g_rows  = (const int*)d_in[5];
  const int* g_cols  = (const int*)d_in[6];
  const int* m1_rows = (const int*)d_in[7];
  const int* m1_cols = (const int*)d_in[8];
  const int* m2_rows = (const int*)d_in[9];
  const int* m2_cols = (const int*)d_in[10];
  const int* users   = (const int*)d_in[11];
  const int* items   = (const int*)d_in[12];

  float* A = (float*)d_ws;  // (NU+NI)*128 f32 = 76.8 MB accumulator

  // 1) A = concat(user_table, item_table)
  long nt4 = (long)(NU + NI) * (DD / 4);
  init_acc<<<(unsigned)((nt4 + 255) / 256), 256, 0, stream>>>(
      (const float4*)user_table, (const float4*)item_table, (float4*)A);

  // 2) graph scatter (2M edges, 32 lanes/edge)
  long tg = (long)NNZ_G * 32;
  scatter_g<<<(unsigned)((tg + 255) / 256), 256, 0, stream>>>(
      g_rows, g_cols, g_vals, user_table, item_table, A, NNZ_G);

  // 3) m1 scatter into user rows, m2 scatter into item rows
  long tm = (long)NNZ_M * 32;
  scatter_m<<<(unsigned)((tm + 255) / 256), 256, 0, stream>>>(
      m1_rows, m1_cols, m1_vals, user_table, A, 0, NNZ_M);
  scatter_m<<<(unsigned)((tm + 255) / 256), 256, 0, stream>>>(
      m2_rows, m2_cols, m2_vals, item_table, A, (long)NU, NNZ_M);

  // 4) 8192 pair dots via f32 WMMA: 512 blocks x 1 wave x 16 pairs
  wmma_pair_dot<<<BQ / 16, 32, 0, stream>>>(A, users, items, (float*)d_out);
}